// ContrastiveLoss_21328807592492
// MI455X (gfx1250) — compile-verified
//
#include <hip/hip_runtime.h>
#include <cstdint>
#include <cstddef>

// ---------------------------------------------------------------------------
// Supervised contrastive loss, MI455X (gfx1250).
//   sim = EN·ENᵀ (B=8192, D=256): 2 passes * 34.4 GFLOP fp32 via
//   V_WMMA_F32_16X16X4_F32 (native f32 WMMA; bf16 rejected: 1/T=14.3x error
//   amplification through exp()).  All tiles staged into LDS via the Tensor
//   Data Mover (tensor_load_to_lds) with descriptor padding giving a
//   conflict-free 260-float LDS row stride.  One wave32 per block => zero
//   barriers; the TDM double-buffer is synchronized with s_wait_tensorcnt.
// ---------------------------------------------------------------------------

#define D          256
#define TILE_N     16
#define LDS_STRIDE 260                    // 256 + 4 (TDM pad: 4 dw per 256 dw)
#define TEMP_INV   14.2857142857142857f   // 1 / 0.07
#define EPSF       1e-8f

typedef float v2f  __attribute__((ext_vector_type(2)));
typedef float v8f  __attribute__((ext_vector_type(8)));
typedef unsigned int u32x4 __attribute__((ext_vector_type(4)));
typedef int   i32x8 __attribute__((ext_vector_type(8)));
typedef int   i32x4 __attribute__((ext_vector_type(4)));
typedef long long ll64;

// ---------------------------------------------------------------------------
// TDM: load a [tile_rows x 256] f32 tile of the row-major [Brows x 256]
// tensor at gptr into LDS at lds_addr, inserting 4 pad DWORDs after every
// 256 DWORDs (=> LDS row stride 260 floats, conflict-free for WMMA reads).
// D# layout per CDNA5 ISA ch.10.8 / CDNA5 ISA.
// This toolchain exposes the 6-arg builtin form (clang-23 / therock-10.0):
//   (u32x4 g0, i32x8 g1, i32x4 g2, i32x4 g3, i32x8 extra, i32 cpol)
// ---------------------------------------------------------------------------
__device__ static inline void tdm_load_tile(unsigned lds_addr,
                                            const float* gptr,
                                            int tile_rows, int Brows)
{
    const uint64_t ga = (uint64_t)(uintptr_t)gptr;

    u32x4 g0;
    g0[0] = 1u;                                        // count=1 valid D#
    g0[1] = lds_addr;                                  // lds_addr [63:32]
    g0[2] = (unsigned)(ga & 0xffffffffull);            // global_addr lo
    g0[3] = (unsigned)((ga >> 32) & 0x01ffffffull)     // global_addr [56:32]
          | 0x80000000u;                               // type=2 ("image")

    i32x8 g1;
    g1[0] = (2 << 16)                                  // data_size = 4 B
          | (1 << 20)                                  // pad_enable
          | (7 << 22)                                  // pad_interval: 256 dw
          | (3 << 25);                                 // pad_amount:   4 dw
    g1[1] = (int)(((unsigned)D & 0xffffu) << 16);      // tensor_dim0 lo16
    g1[2] = (int)((((unsigned)D >> 16) & 0xffffu)      // tensor_dim0 hi16
          | (((unsigned)Brows & 0xffffu) << 16));      // tensor_dim1 lo16
    g1[3] = (int)((((unsigned)Brows >> 16) & 0xffffu)  // tensor_dim1 hi16
          | (((unsigned)D & 0xffffu) << 16));          // tile_dim0 = 256
    g1[4] = tile_rows & 0xffff;                        // tile_dim1 (tile_dim2=0)
    g1[5] = D;                                         // tensor_dim0_stride lo32
    g1[6] = 0;                                         // stride hi / dim1_stride
    g1[7] = 0;

    i32x4 z4 = {0, 0, 0, 0};
    i32x8 z8 = {0, 0, 0, 0, 0, 0, 0, 0};
    __builtin_amdgcn_tensor_load_to_lds(g0, g1, z4, z4, z8, 0);
}

__device__ static inline void tdm_wait(int pending)
{
    if (pending) __builtin_amdgcn_s_wait_tensorcnt(1);
    else         __builtin_amdgcn_s_wait_tensorcnt(0);
    asm volatile("" ::: "memory");   // keep ds_loads below the wait
}

// ---------------------------------------------------------------------------
// Kernel 1: row L2-normalize.  One wave32 per row (D=256 -> 8 floats/lane).
// ---------------------------------------------------------------------------
__global__ __launch_bounds__(256) void cl_normalize_kernel(
    const float* __restrict__ x, float* __restrict__ en, int B)
{
    int wave = (int)((blockIdx.x * blockDim.x + threadIdx.x) >> 5);
    int lane = threadIdx.x & 31;
    if (wave >= B) return;

    const float* row = x + (size_t)wave * D;
    float4 a = *(const float4*)(row + lane * 8);
    float4 b = *(const float4*)(row + lane * 8 + 4);
    float ss = a.x*a.x + a.y*a.y + a.z*a.z + a.w*a.w
             + b.x*b.x + b.y*b.y + b.z*b.z + b.w*b.w;
    #pragma unroll
    for (int off = 16; off > 0; off >>= 1)
        ss += __shfl_xor(ss, off, 32);

    float inv = 1.0f / fmaxf(sqrtf(ss), 1e-12f);
    a.x *= inv; a.y *= inv; a.z *= inv; a.w *= inv;
    b.x *= inv; b.y *= inv; b.z *= inv; b.w *= inv;
    float* orow = en + (size_t)wave * D;
    *(float4*)(orow + lane * 8)     = a;
    *(float4*)(orow + lane * 8 + 4) = b;
}

// ---------------------------------------------------------------------------
// Kernel 2: fused sim + masked loss.  One wave = 16 rows; 512 column tiles
// TDM-double-buffered through LDS; two passes (neg_sum, then positives).
// ---------------------------------------------------------------------------
__global__ __launch_bounds__(32) void cl_loss_kernel(
    const float* __restrict__ en, const ll64* __restrict__ labels,
    float2* __restrict__ partials, int B)
{
    __shared__ __align__(16) float Abuf[TILE_N][LDS_STRIDE];
    __shared__ __align__(16) float Bbuf[2][TILE_N][LDS_STRIDE];

    const int lane  = threadIdx.x & 31;
    const int half  = lane >> 4;         // 0: rows e, 1: rows e+8
    const int ln    = lane & 15;
    const int rbase = (int)blockIdx.x * TILE_N;

    const unsigned ldsA  = (unsigned)(uintptr_t)&Abuf[0][0];
    const unsigned ldsB0 = (unsigned)(uintptr_t)&Bbuf[0][0][0];
    const unsigned ldsB1 = (unsigned)(uintptr_t)&Bbuf[1][0][0];

    // prologue: async-stage A tile and first B tile (TENSORcnt in-order)
    tdm_load_tile(ldsA,  en + (size_t)rbase * D, TILE_N, B);
    tdm_load_tile(ldsB0, en,                     TILE_N, B);

    // per-lane row metadata: this lane owns rows e + 8*half of the tile
    int rlab[8], rgid[8];
    #pragma unroll
    for (int e = 0; e < 8; ++e) {
        rgid[e] = rbase + e + 8 * half;
        rlab[e] = (int)labels[rgid[e]];
    }

    const int ntiles = B / TILE_N;

    // =========================== PHASE 1 ==============================
    float negsum[8]; int negcnt[8];
    #pragma unroll
    for (int e = 0; e < 8; ++e) { negsum[e] = 0.0f; negcnt[e] = 0; }

    for (int c = 0; c < ntiles; ++c) {
        if (c + 1 < ntiles)
            tdm_load_tile((((c + 1) & 1) ? ldsB1 : ldsB0),
                          en + (size_t)(c + 1) * TILE_N * D, TILE_N, B);
        tdm_wait(c + 1 < ntiles);        // tile c landed; c+1 may be in flight

        const int cbase = c * TILE_N;
        const int clab  = (int)labels[cbase + ln];
        const float* bb = &Bbuf[c & 1][0][0];

        v8f acc = {};
        #pragma unroll 8
        for (int k0 = 0; k0 < D; k0 += 4) {
            v2f a = *(const v2f*)&Abuf[ln][k0 + 2 * half];
            v2f b = *(const v2f*)(bb + ln * LDS_STRIDE + k0 + 2 * half);
            acc = __builtin_amdgcn_wmma_f32_16x16x4_f32(
                false, a, false, b, (short)0, acc, false, false);
        }

        #pragma unroll
        for (int e = 0; e < 8; ++e) {    // branchless masked accumulate
            float s  = acc[e] * TEMP_INV;
            float es = expf(s);
            bool neg = (rlab[e] != clab);          // != label => off-diagonal
            negsum[e] += neg ? es : 0.0f;
            negcnt[e] += neg ? 1 : 0;
        }
    }

    // butterfly over the 16 lanes sharing each row set
    #pragma unroll
    for (int e = 0; e < 8; ++e) {
        #pragma unroll
        for (int off = 8; off > 0; off >>= 1) {
            negsum[e] += __shfl_xor(negsum[e], off, 16);
            negcnt[e] += __shfl_xor(negcnt[e], off, 16);
        }
    }
    float Crow[8];
    #pragma unroll
    for (int e = 0; e < 8; ++e) Crow[e] = negsum[e] + EPSF;

    // =========================== PHASE 2 ==============================
    float possum[8]; int poscnt[8];
    #pragma unroll
    for (int e = 0; e < 8; ++e) { possum[e] = 0.0f; poscnt[e] = 0; }

    tdm_load_tile(ldsB0, en, TILE_N, B);

    for (int c = 0; c < ntiles; ++c) {
        if (c + 1 < ntiles)
            tdm_load_tile((((c + 1) & 1) ? ldsB1 : ldsB0),
                          en + (size_t)(c + 1) * TILE_N * D, TILE_N, B);
        tdm_wait(c + 1 < ntiles);

        const int cbase = c * TILE_N;
        const int clab  = (int)labels[cbase + ln];
        const int cgid  = cbase + ln;
        const float* bb = &Bbuf[c & 1][0][0];

        v8f acc = {};
        #pragma unroll 8
        for (int k0 = 0; k0 < D; k0 += 4) {
            v2f a = *(const v2f*)&Abuf[ln][k0 + 2 * half];
            v2f b = *(const v2f*)(bb + ln * LDS_STRIDE + k0 + 2 * half);
            acc = __builtin_amdgcn_wmma_f32_16x16x4_f32(
                false, a, false, b, (short)0, acc, false, false);
        }

        #pragma unroll
        for (int e = 0; e < 8; ++e) {
            float s   = acc[e] * TEMP_INV;
            float per = logf(expf(s) + Crow[e]) - s;   // -log(e^s/(e^s+C))
            bool pos  = (rlab[e] == clab) & (rgid[e] != cgid);
            possum[e] += pos ? per : 0.0f;
            poscnt[e] += pos ? 1 : 0;
        }
    }

    #pragma unroll
    for (int e = 0; e < 8; ++e) {
        #pragma unroll
        for (int off = 8; off > 0; off >>= 1) {
            possum[e] += __shfl_xor(possum[e], off, 16);
            poscnt[e] += __shfl_xor(poscnt[e], off, 16);
        }
    }

    // fold 8 rows per half (row_valid gating), merge halves via xor-16
    float t = 0.0f, cnt = 0.0f;
    #pragma unroll
    for (int e = 0; e < 8; ++e) {
        bool valid = (poscnt[e] > 0) & (negcnt[e] > 0);
        t   += valid ? possum[e]        : 0.0f;
        cnt += valid ? (float)poscnt[e] : 0.0f;
    }
    t   += __shfl_xor(t,   16, 32);
    cnt += __shfl_xor(cnt, 16, 32);
    if (lane == 0) partials[blockIdx.x] = make_float2(t, cnt);
}

// ---------------------------------------------------------------------------
// Kernel 3: deterministic final fold of per-block partials.
// ---------------------------------------------------------------------------
__global__ void cl_finalize_kernel(const float2* __restrict__ partials,
                                   int n, float* __restrict__ out)
{
    if (threadIdx.x == 0 && blockIdx.x == 0) {
        float t = 0.0f, c = 0.0f;
        for (int i = 0; i < n; ++i) { t += partials[i].x; c += partials[i].y; }
        out[0] = (c > 0.0f) ? (t / fmaxf(c, 1.0f)) : 0.0f;
    }
}

// ---------------------------------------------------------------------------
extern "C" void kernel_launch(void* const* d_in, const int* in_sizes, int n_in,
                              void* d_out, int out_size, void* d_ws, size_t ws_size,
                              hipStream_t stream)
{
    const float* x      = (const float*)d_in[0];
    const ll64*  labels = (const ll64*)d_in[1];
    float*       out    = (float*)d_out;

    const int B = in_sizes[1];                  // 8192
    float*  en       = (float*)d_ws;            // B*D normalized embeddings
    float2* partials = (float2*)((char*)d_ws + (size_t)B * D * sizeof(float));

    const int nblocks = B / TILE_N;             // 512

    cl_normalize_kernel<<<(B + 7) / 8, 256, 0, stream>>>(x, en, B);
    cl_loss_kernel<<<nblocks, 32, 0, stream>>>(en, labels, partials, B);
    cl_finalize_kernel<<<1, 32, 0, stream>>>(partials, nblocks, out);
}